// FNOBlock2D_30090540876185
// MI455X (gfx1250) — compile-verified
//
#include <hip/hip_runtime.h>
#include <math.h>

// FNO block: B=16, H=W=128, C=64, DEPTH=4, modes 16x16 (rows 0..15 & 112..127)
#define NB 16
#define NH 128
#define NW 128
#define NC 64
#define NM 16
#define NKX 32
#define FNO_DEPTH 4

typedef __attribute__((ext_vector_type(2))) float v2f;
typedef __attribute__((ext_vector_type(8))) float v8f;

__device__ __forceinline__ v8f wmma4(v2f a, v2f b, v8f c) {
  // V_WMMA_F32_16X16X4_F32 : D = A(16x4) * B(4x16) + C, full f32
  return __builtin_amdgcn_wmma_f32_16x16x4_f32(false, a, false, b, (short)0, c,
                                               false, false);
}

__device__ __forceinline__ float gelu_tanh(float x) {
  const float k = 0.7978845608028654f;  // sqrt(2/pi)
  return 0.5f * x * (1.0f + tanhf(k * (x + 0.044715f * x * x * x)));
}

// ---------------------------------------------------------------------------
// Precompute truncated-DFT factor matrices (deterministic, tiny).
//  F1[32][128] : forward W-rDFT.  rows 0..15 = cos, 16..31 = -sin
//  F2[64][256] : forward H-DFT on (Yre||Yim), rows = (kx,part)
//  F3[256][64] : inverse H-DFT -> (Ure rows 0..127, Uim rows 128..255)
//  G [128][32] : irfft along W (cols = (part,ky)), includes 1/(H*W) and x2
// ---------------------------------------------------------------------------
__global__ __launch_bounds__(256) void init_dft(float* __restrict__ F1,
                                                float* __restrict__ F2,
                                                float* __restrict__ F3,
                                                float* __restrict__ G) {
  const float TP = 6.283185307179586f;
  int id = blockIdx.x * 256 + threadIdx.x;
  if (id < 4096) {  // F1
    int r = id >> 7, w = id & 127;
    float th = TP * (float)((r & 15) * w) / 128.0f;
    F1[id] = (r < 16) ? cosf(th) : -sinf(th);
  } else if (id < 4096 + 16384) {  // F2
    int j = id - 4096;
    int m = j >> 8, k = j & 255;
    int kx = m >> 1, part = m & 1;
    int rr = (kx < NM) ? kx : 96 + kx;  // true row index (112..127)
    int h = k & 127, isIm = k >> 7;
    float th = TP * (float)(rr * h) / 128.0f;
    float c = cosf(th), s = sinf(th);
    F2[j] = (part == 0) ? (isIm ? s : c) : (isIm ? c : -s);
  } else if (id < 4096 + 16384 + 16384) {  // F3
    int j = id - 20480;
    int m = j >> 6, k = j & 63;
    int ph = m >> 7, h = m & 127;
    int kx = k >> 1, pk = k & 1;
    int rr = (kx < NM) ? kx : 96 + kx;
    float th = TP * (float)(rr * h) / 128.0f;  // e^{+i th}
    float c = cosf(th), s = sinf(th);
    F3[j] = (ph == 0) ? (pk == 0 ? c : -s) : (pk == 0 ? s : c);
  } else if (id < 40960) {  // G
    int j = id - 36864;
    int w = j >> 5, cidx = j & 31;
    int part = cidx >> 4, ky = cidx & 15;
    float th = TP * (float)(ky * w) / 128.0f;
    float sc = ((ky == 0) ? 1.0f : 2.0f) / 16384.0f;  // 1/(H*W) folded here
    G[j] = (part == 0) ? sc * cosf(th) : -sc * sinf(th);
  }
}

// ---------------------------------------------------------------------------
// Repack complex weights into real block matrices:
//  Wrep[kx(32)][ky(16)][i2(128)][o2(128)] = [[ Wre  Wim ],
//                                            [-Wim  Wre ]]
// (kx<16 -> w1, kx>=16 -> w2)
// ---------------------------------------------------------------------------
__global__ __launch_bounds__(256) void repack_w(
    const float* __restrict__ w1r, const float* __restrict__ w1i,
    const float* __restrict__ w2r, const float* __restrict__ w2i,
    float* __restrict__ Wrep, int d) {
  int idx = blockIdx.x * 256 + threadIdx.x;  // 512*128*128
  int o2 = idx & 127;
  int i2 = (idx >> 7) & 127;
  int ky = (idx >> 14) & 15;
  int kx = idx >> 18;
  int i = i2 & 63, o = o2 & 63;
  const float *wr, *wi;
  int kk;
  if (kx < NM) { wr = w1r; wi = w1i; kk = kx; }
  else         { wr = w2r; wi = w2i; kk = kx - NM; }
  int widx = (((d * NC + i) * NC + o) * NM + kk) * NM + ky;
  float re = wr[widx], im = wi[widx];
  Wrep[idx] = (i2 < 64) ? ((o2 < 64) ? re : im) : ((o2 < 64) ? -im : re);
}

// ---------------------------------------------------------------------------
// Stage A: Y[b][ky2(32)][h][i] = F1(32x128) x X[b][h](128x64)   batch B*H
// ---------------------------------------------------------------------------
__global__ __launch_bounds__(256) void stageA(const float* __restrict__ x,
                                              const float* __restrict__ F1,
                                              float* __restrict__ Y) {
  extern __shared__ float smem[];
  float* sX = smem;         // 8192
  float* sF = smem + 8192;  // 4096
  int bh = blockIdx.x;
  int b = bh >> 7, h = bh & 127;
  int t = threadIdx.x;
  {
    const float4* s4 = (const float4*)(x + (size_t)bh * (NW * NC));
    float4* d4 = (float4*)sX;
    for (int j = t; j < 2048; j += 256) d4[j] = s4[j];
    const float4* f4 = (const float4*)F1;
    float4* g4 = (float4*)sF;
    for (int j = t; j < 1024; j += 256) g4[j] = f4[j];
  }
  __syncthreads();
  int wave = t >> 5, lane = t & 31;
  int half = lane >> 4, mn = lane & 15;
  int tm = (wave >> 2) * 16;  // M=32 -> 2 tile rows
  int tn = (wave & 3) * 16;   // N=64 -> 4 tile cols
  v8f acc = {0.f, 0.f, 0.f, 0.f, 0.f, 0.f, 0.f, 0.f};
#pragma unroll 4
  for (int k0 = 0; k0 < 128; k0 += 4) {
    int ka = k0 + 2 * half;
    v2f a, bb;
    a.x = sF[(tm + mn) * 128 + ka];
    a.y = sF[(tm + mn) * 128 + ka + 1];
    bb.x = sX[ka * 64 + tn + mn];
    bb.y = sX[(ka + 1) * 64 + tn + mn];
    acc = wmma4(a, bb, acc);
  }
#pragma unroll
  for (int r = 0; r < 8; ++r) {
    int ky2 = tm + r + 8 * half;
    Y[(((size_t)b * NKX + ky2) * NH + h) * NC + (tn + mn)] = acc[r];
  }
}

// ---------------------------------------------------------------------------
// Stage B: Z[kx][ky][part][b][i] : per (b,ky)  F2(64x256) x [Yre;Yim](256x64)
// ---------------------------------------------------------------------------
__global__ __launch_bounds__(256) void stageB(const float* __restrict__ Y,
                                              const float* __restrict__ F2,
                                              float* __restrict__ Z) {
  extern __shared__ float smem[];
  float* sF = smem;          // 16384
  float* sY = smem + 16384;  // 16384
  int bk = blockIdx.x;
  int b = bk >> 4, ky = bk & 15;
  int t = threadIdx.x;
  {
    const float4* y0 = (const float4*)(Y + ((size_t)b * NKX + ky) * NH * NC);
    const float4* y1 =
        (const float4*)(Y + ((size_t)b * NKX + NM + ky) * NH * NC);
    float4* d4 = (float4*)sY;
    for (int j = t; j < 2048; j += 256) d4[j] = y0[j];
    for (int j = t; j < 2048; j += 256) d4[2048 + j] = y1[j];
    const float4* f4 = (const float4*)F2;
    float4* g4 = (float4*)sF;
    for (int j = t; j < 4096; j += 256) g4[j] = f4[j];
  }
  __syncthreads();
  int wave = t >> 5, lane = t & 31;
  int half = lane >> 4, mn = lane & 15;
  for (int it = 0; it < 2; ++it) {
    int tile = wave * 2 + it;   // 16 tiles (4x4)
    int tm = (tile >> 2) * 16;
    int tn = (tile & 3) * 16;
    v8f acc = {0.f, 0.f, 0.f, 0.f, 0.f, 0.f, 0.f, 0.f};
#pragma unroll 4
    for (int k0 = 0; k0 < 256; k0 += 4) {
      int ka = k0 + 2 * half;
      v2f a, bb;
      a.x = sF[(tm + mn) * 256 + ka];
      a.y = sF[(tm + mn) * 256 + ka + 1];
      bb.x = sY[ka * 64 + tn + mn];
      bb.y = sY[(ka + 1) * 64 + tn + mn];
      acc = wmma4(a, bb, acc);
    }
#pragma unroll
    for (int r = 0; r < 8; ++r) {
      int m = tm + r + 8 * half;
      int kx = m >> 1, part = m & 1;
      Z[((((size_t)kx * NM + ky) * 2 + part) * NB + b) * NC + (tn + mn)] =
          acc[r];
    }
  }
}

// ---------------------------------------------------------------------------
// Stage C: per mode (kx,ky): Out(16x128) = Zcat(16x128) x Wrep(128x128)
//  writes Z2[b][ky][k2(64)][o(64)] with k2 = 2*kx + (re/im)
// ---------------------------------------------------------------------------
__global__ __launch_bounds__(256) void stageC(const float* __restrict__ Z,
                                              const float* __restrict__ Wrep,
                                              float* __restrict__ Z2) {
  extern __shared__ float smem[];
  float* sW = smem;          // 16384
  float* sZ = smem + 16384;  // 2048 : [part][b][i]
  int mode = blockIdx.x;
  int kx = mode >> 4, ky = mode & 15;
  int t = threadIdx.x;
  {
    const float4* w4 = (const float4*)(Wrep + (size_t)mode * 16384);
    float4* d4 = (float4*)sW;
    for (int j = t; j < 4096; j += 256) d4[j] = w4[j];
    const float4* z4 = (const float4*)(Z + (size_t)mode * 2048);
    float4* e4 = (float4*)sZ;
    for (int j = t; j < 512; j += 256) e4[j] = z4[j];
  }
  __syncthreads();
  int wave = t >> 5, lane = t & 31;
  int half = lane >> 4, mn = lane & 15;
  int tn = wave * 16;  // N=128 -> 8 tiles, one per wave
  v8f acc = {0.f, 0.f, 0.f, 0.f, 0.f, 0.f, 0.f, 0.f};
#pragma unroll 4
  for (int k0 = 0; k0 < 128; k0 += 4) {
    int ka = k0 + 2 * half;
    v2f a, bb;
    a.x = sZ[(ka >> 6) * 1024 + mn * 64 + (ka & 63)];
    a.y = sZ[((ka + 1) >> 6) * 1024 + mn * 64 + ((ka + 1) & 63)];
    bb.x = sW[ka * 128 + tn + mn];
    bb.y = sW[(ka + 1) * 128 + tn + mn];
    acc = wmma4(a, bb, acc);
  }
#pragma unroll
  for (int r = 0; r < 8; ++r) {
    int bm = r + 8 * half;  // batch index b
    int o2 = tn + mn;
    int krow = 2 * kx + (o2 >> 6);
    Z2[(((size_t)bm * NM + ky) * 64 + krow) * NC + (o2 & 63)] = acc[r];
  }
}

// ---------------------------------------------------------------------------
// Stage D: per (b,ky): U(256x64) = F3(256x64) x Z2blk(64x64)
//  writes U[b][h][part][ky][o]
// ---------------------------------------------------------------------------
__global__ __launch_bounds__(256) void stageD(const float* __restrict__ Z2,
                                              const float* __restrict__ F3,
                                              float* __restrict__ U) {
  extern __shared__ float smem[];
  float* sF = smem;          // 16384
  float* sB = smem + 16384;  // 4096
  int bk = blockIdx.x;
  int b = bk >> 4, ky = bk & 15;
  int t = threadIdx.x;
  {
    const float4* f4 = (const float4*)F3;
    float4* d4 = (float4*)sF;
    for (int j = t; j < 4096; j += 256) d4[j] = f4[j];
    const float4* z4 = (const float4*)(Z2 + ((size_t)b * NM + ky) * 4096);
    float4* e4 = (float4*)sB;
    for (int j = t; j < 1024; j += 256) e4[j] = z4[j];
  }
  __syncthreads();
  int wave = t >> 5, lane = t & 31;
  int half = lane >> 4, mn = lane & 15;
  for (int it = 0; it < 8; ++it) {
    int tile = wave * 8 + it;   // 64 tiles (16x4)
    int tm = (tile >> 2) * 16;
    int tn = (tile & 3) * 16;
    v8f acc = {0.f, 0.f, 0.f, 0.f, 0.f, 0.f, 0.f, 0.f};
#pragma unroll 4
    for (int k0 = 0; k0 < 64; k0 += 4) {
      int ka = k0 + 2 * half;
      v2f a, bb;
      a.x = sF[(tm + mn) * 64 + ka];
      a.y = sF[(tm + mn) * 64 + ka + 1];
      bb.x = sB[ka * 64 + tn + mn];
      bb.y = sB[(ka + 1) * 64 + tn + mn];
      acc = wmma4(a, bb, acc);
    }
#pragma unroll
    for (int r = 0; r < 8; ++r) {
      int m = tm + r + 8 * half;  // h2
      int part = m >> 7, h = m & 127;
      U[((((size_t)b * NH + h) * 2 + part) * NM + ky) * NC + (tn + mn)] =
          acc[r];
    }
  }
}

// ---------------------------------------------------------------------------
// Stage E (fused): per (b,h):
//   out = gelu( G(128x32) x Ucat(32x64)  +  Xrow(128x64) x CK(64x64) + cb )
// ---------------------------------------------------------------------------
__global__ __launch_bounds__(256) void stageE(
    const float* __restrict__ xin, const float* __restrict__ U,
    const float* __restrict__ G, const float* __restrict__ ck,
    const float* __restrict__ cb, float* __restrict__ xout, int d) {
  extern __shared__ float smem[];
  float* sU = smem;          // 2048
  float* sX = smem + 2048;   // 8192
  float* sC = smem + 10240;  // 4096
  float* sG = smem + 14336;  // 4096
  float* sb = smem + 18432;  // 64
  int bh = blockIdx.x;
  int t = threadIdx.x;
  {
    const float4* u4 = (const float4*)(U + (size_t)bh * 2048);
    float4* a4 = (float4*)sU;
    for (int j = t; j < 512; j += 256) a4[j] = u4[j];
    const float4* x4 = (const float4*)(xin + (size_t)bh * (NW * NC));
    float4* b4 = (float4*)sX;
    for (int j = t; j < 2048; j += 256) b4[j] = x4[j];
    const float4* c4 = (const float4*)(ck + (size_t)d * 4096);
    float4* cc4 = (float4*)sC;
    for (int j = t; j < 1024; j += 256) cc4[j] = c4[j];
    const float4* g4 = (const float4*)G;
    float4* gg4 = (float4*)sG;
    for (int j = t; j < 1024; j += 256) gg4[j] = g4[j];
    const float4* bb4 = (const float4*)(cb + (size_t)d * 64);
    if (t < 16) ((float4*)sb)[t] = bb4[t];
  }
  __syncthreads();
  int wave = t >> 5, lane = t & 31;
  int half = lane >> 4, mn = lane & 15;
  for (int it = 0; it < 4; ++it) {
    int tile = wave * 4 + it;   // 32 tiles (8x4)
    int tm = (tile >> 2) * 16;
    int tn = (tile & 3) * 16;
    v8f acc = {0.f, 0.f, 0.f, 0.f, 0.f, 0.f, 0.f, 0.f};
    // irfft along W : K = 32
#pragma unroll
    for (int k0 = 0; k0 < 32; k0 += 4) {
      int ka = k0 + 2 * half;
      v2f a, bb;
      a.x = sG[(tm + mn) * 32 + ka];
      a.y = sG[(tm + mn) * 32 + ka + 1];
      bb.x = sU[ka * 64 + tn + mn];
      bb.y = sU[(ka + 1) * 64 + tn + mn];
      acc = wmma4(a, bb, acc);
    }
    // 1x1 conv : K = 64 (same accumulator)
#pragma unroll 4
    for (int k0 = 0; k0 < 64; k0 += 4) {
      int ka = k0 + 2 * half;
      v2f a, bb;
      a.x = sX[(tm + mn) * 64 + ka];
      a.y = sX[(tm + mn) * 64 + ka + 1];
      bb.x = sC[ka * 64 + tn + mn];
      bb.y = sC[(ka + 1) * 64 + tn + mn];
      acc = wmma4(a, bb, acc);
    }
#pragma unroll
    for (int r = 0; r < 8; ++r) {
      int w = tm + r + 8 * half;
      int o = tn + mn;
      float v = acc[r] + sb[o];
      xout[(size_t)bh * (NW * NC) + (size_t)w * NC + o] = gelu_tanh(v);
    }
  }
}

// ---------------------------------------------------------------------------
extern "C" void kernel_launch(void* const* d_in, const int* in_sizes, int n_in,
                              void* d_out, int out_size, void* d_ws,
                              size_t ws_size, hipStream_t stream) {
  const float* x   = (const float*)d_in[0];
  const float* w1r = (const float*)d_in[1];
  const float* w1i = (const float*)d_in[2];
  const float* w2r = (const float*)d_in[3];
  const float* w2i = (const float*)d_in[4];
  const float* ck  = (const float*)d_in[5];
  const float* cb  = (const float*)d_in[6];
  float* out = (float*)d_out;
  float* ws = (float*)d_ws;

  const size_t NXY = (size_t)NB * NH * NW * NC;  // 16,777,216
  float* xA = ws;
  float* xB = xA + NXY;
  float* Y  = xB + NXY;                            // 4,194,304
  float* Z  = Y + (size_t)NB * NKX * NH * NC;      // 1,048,576
  float* Z2 = Z + (size_t)NKX * NM * 2 * NB * NC;  // 1,048,576
  float* U  = Z2 + (size_t)NB * NM * 64 * NC;      // 4,194,304
  float* Wrep = U + (size_t)NB * NH * 2 * NM * NC; // 8,388,608
  float* F1 = Wrep + (size_t)NKX * NM * 128 * 128;
  float* F2 = F1 + 4096;
  float* F3 = F2 + 16384;
  float* G  = F3 + 16384;

  init_dft<<<160, 256, 0, stream>>>(F1, F2, F3, G);

  const float* cur = x;
  for (int d = 0; d < FNO_DEPTH; ++d) {
    float* nxt = (d == FNO_DEPTH - 1) ? out : ((d & 1) ? xB : xA);
    repack_w<<<32768, 256, 0, stream>>>(w1r, w1i, w2r, w2i, Wrep, d);
    stageA<<<NB * NH, 256, (8192 + 4096) * sizeof(float), stream>>>(cur, F1, Y);
    stageB<<<NB * NM, 256, (16384 + 16384) * sizeof(float), stream>>>(Y, F2, Z);
    stageC<<<NKX * NM, 256, (16384 + 2048) * sizeof(float), stream>>>(Z, Wrep,
                                                                     Z2);
    stageD<<<NB * NM, 256, (16384 + 4096) * sizeof(float), stream>>>(Z2, F3, U);
    stageE<<<NB * NH, 256, (2048 + 8192 + 4096 + 4096 + 64) * sizeof(float),
             stream>>>(cur, U, G, ck, cb, nxt, d);
    cur = nxt;
  }
  (void)in_sizes; (void)n_in; (void)out_size; (void)ws_size;
}